// tbspp_69114613729375
// MI455X (gfx1250) — compile-verified
//
#include <hip/hip_runtime.h>
#include <hip/hip_bf16.h>

typedef __attribute__((ext_vector_type(2))) float v2f;
typedef __attribute__((ext_vector_type(8))) float v8f;

#define EMB 256
#define NNODES 1024
#define NB 32
#define MC 8
#define OUT1 240
#define OUT2 120
#define LABEL 104
#define MTOT (NB * NNODES)   // 32768 rows

// ---------------------------------------------------------------------------
// Kernel 1: tree convolution feature builder.
// One wave (32 lanes) per node. Produces X0[g, 3*e + k], k in {t,r,l}.
// Children are gathered from batch 0's lookup (reference quirk), node 0 -> 0.
// ---------------------------------------------------------------------------
__global__ __launch_bounds__(256) void tree_conv_kernel(
    const float* __restrict__ nodes, const int* __restrict__ children,
    float* __restrict__ X0) {
  const int wid = threadIdx.x >> 5;
  const int lane = threadIdx.x & 31;
  const int g = blockIdx.x * 8 + wid;          // node id in [0, 32768)
  const int b = g >> 10;
  const int n = g & (NNODES - 1);

  const float* __restrict__ parent = nodes + ((size_t)b * NNODES + n) * EMB;
  const int* __restrict__ ch = children + ((size_t)b * NNODES + n) * MC;

  int idx[MC];
  int ns = 0;
#pragma unroll
  for (int j = 0; j < MC; ++j) {
    idx[j] = ch[j];
    ns += (idx[j] != 0) ? 1 : 0;
  }
  const float inv = (ns > 1) ? 1.0f / (float)(ns - 1) : 0.0f;
  float cr[MC], cl[MC];
#pragma unroll
  for (int j = 0; j < MC; ++j) {
    const float mask = (idx[j] != 0) ? 1.0f : 0.0f;
    float r;
    if (ns == 1)
      r = (j == 0) ? 0.5f : 0.0f;          // "singles" path of reference
    else
      r = (float)j * mask * inv;           // child_idx / (ns-1)
    cr[j] = r;
    cl[j] = (1.0f - r) * mask;
  }

  float* __restrict__ out = X0 + (size_t)g * (3 * EMB);
#pragma unroll
  for (int t = 0; t < EMB / 32; ++t) {
    const int e = lane + t * 32;
    const float p = parent[e];
    float sr = 0.0f, sl = 0.0f;
#pragma unroll
    for (int j = 0; j < MC; ++j) {
      if (idx[j] != 0) {
        // lookup[0][idx] == nodes[0, idx, :] for idx >= 1
        const float v = nodes[(size_t)idx[j] * EMB + e];
        sr += cr[j] * v;
        sl += cl[j] * v;
      }
    }
    out[3 * e + 0] = p;
    out[3 * e + 1] = sr;
    out[3 * e + 2] = sl;
  }
}

// ---------------------------------------------------------------------------
// Kernel 2/3: C = tanh(A[M,K] * W[N,K]^T + bias), f32 WMMA 16x16x4.
// Block: 256 threads (8 waves), BM=128 rows, all NTILES column tiles.
// Wave w owns M-subtile w; A fragment reused across all N tiles.
// Register-prefetch double buffering: next K-slice's global loads are issued
// before the WMMA chain so they complete behind 60 matrix ops.
// B fragments for all N tiles are pulled from LDS into registers first, so
// the 15 WMMAs issue back-to-back behind a single dscnt wait.
// ---------------------------------------------------------------------------
template <int NTILES>
__global__ __launch_bounds__(256) void gemm_bias_tanh_kernel(
    const float* __restrict__ A, const float* __restrict__ W,
    const float* __restrict__ bias, float* __restrict__ C, int N, int K) {
  __shared__ float As[128][17];              // +1 pad: conflict-free frag reads
  __shared__ float Bs[16][NTILES * 16];

  constexpr int BF4 = (NTILES * 64 + 255) / 256;  // B float4s per thread

  const int tid = threadIdx.x;
  const int wid = tid >> 5;
  const int lane = tid & 31;
  const int lane15 = lane & 15;
  const int khalf = lane >> 4;
  const int m0 = blockIdx.x * 128;
  const int ml = wid * 16 + lane15;

  v8f acc[NTILES];
#pragma unroll
  for (int nt = 0; nt < NTILES; ++nt) acc[nt] = (v8f){0, 0, 0, 0, 0, 0, 0, 0};

  float4 pa[2];       // A prefetch: 128x16 tile = 512 float4 / 256 threads
  float4 pb[BF4];     // B prefetch

  // Prologue: fetch K-slice 0 into registers.
  {
#pragma unroll
    for (int i = 0; i < 2; ++i) {
      const int f = tid + i * 256;
      const int row = f >> 2, c4 = (f & 3) * 4;
      pa[i] = *(const float4*)(A + (size_t)(m0 + row) * K + c4);
    }
#pragma unroll
    for (int i = 0; i < BF4; ++i) {
      const int f = tid + i * 256;
      float4 v = make_float4(0.f, 0.f, 0.f, 0.f);
      if (f < NTILES * 64) {
        const int nrow = f >> 2, c4 = (f & 3) * 4;
        if (nrow < N) v = *(const float4*)(W + (size_t)nrow * K + c4);
      }
      pb[i] = v;
    }
  }

  const int kTiles = K / 16;
  for (int t = 0; t < kTiles; ++t) {
    // Commit prefetched slice to LDS.
#pragma unroll
    for (int i = 0; i < 2; ++i) {
      const int f = tid + i * 256;
      const int row = f >> 2, c4 = (f & 3) * 4;
      As[row][c4 + 0] = pa[i].x;
      As[row][c4 + 1] = pa[i].y;
      As[row][c4 + 2] = pa[i].z;
      As[row][c4 + 3] = pa[i].w;
    }
#pragma unroll
    for (int i = 0; i < BF4; ++i) {
      const int f = tid + i * 256;
      if (f < NTILES * 64) {
        const int nrow = f >> 2, c4 = (f & 3) * 4;
        Bs[c4 + 0][nrow] = pb[i].x;
        Bs[c4 + 1][nrow] = pb[i].y;
        Bs[c4 + 2][nrow] = pb[i].z;
        Bs[c4 + 3][nrow] = pb[i].w;
      }
    }
    __syncthreads();

    // Issue next slice's global loads; they retire behind the WMMA chain.
    if (t + 1 < kTiles) {
      const int kt = (t + 1) * 16;
#pragma unroll
      for (int i = 0; i < 2; ++i) {
        const int f = tid + i * 256;
        const int row = f >> 2, c4 = (f & 3) * 4;
        pa[i] = *(const float4*)(A + (size_t)(m0 + row) * K + kt + c4);
      }
#pragma unroll
      for (int i = 0; i < BF4; ++i) {
        const int f = tid + i * 256;
        float4 v = make_float4(0.f, 0.f, 0.f, 0.f);
        if (f < NTILES * 64) {
          const int nrow = f >> 2, c4 = (f & 3) * 4;
          if (nrow < N) v = *(const float4*)(W + (size_t)nrow * K + kt + c4);
        }
        pb[i] = v;
      }
    }

    // 4 K-steps of 16x16x4; A frag reused across all N tiles.
#pragma unroll
    for (int ks = 0; ks < 4; ++ks) {
      const int kb = ks * 4 + khalf * 2;
      v2f a;
      a.x = As[ml][kb + 0];
      a.y = As[ml][kb + 1];
      // Pull all B fragments first so the WMMAs issue back-to-back.
      v2f bf[NTILES];
#pragma unroll
      for (int nt = 0; nt < NTILES; ++nt) {
        bf[nt].x = Bs[kb + 0][nt * 16 + lane15];
        bf[nt].y = Bs[kb + 1][nt * 16 + lane15];
      }
#pragma unroll
      for (int nt = 0; nt < NTILES; ++nt) {
        acc[nt] = __builtin_amdgcn_wmma_f32_16x16x4_f32(
            false, a, false, bf[nt], (short)0, acc[nt], false, false);
      }
    }
    __syncthreads();
  }

  // Epilogue: C/D layout — VGPR r holds rows (r, r+8) across half-waves
#pragma unroll
  for (int nt = 0; nt < NTILES; ++nt) {
    const int n = nt * 16 + lane15;
    if (n < N) {
      const float bv = bias[n];
#pragma unroll
      for (int r = 0; r < 8; ++r) {
        const int m = m0 + wid * 16 + r + khalf * 8;
        C[(size_t)m * N + n] = tanhf(acc[nt][r] + bv);
      }
    }
  }
}

// ---------------------------------------------------------------------------
// Kernel 4: pyramid max pooling. One block per batch, thread = channel.
// Output layout per batch: [k=1: c] [k=2: c*2+i] [k=4: c*4+i] [k=8: c*8+i]
// at offsets 0 / 120 / 360 / 840 (total 1800).
// ---------------------------------------------------------------------------
__global__ __launch_bounds__(128) void pyramid_pool_kernel(
    const float* __restrict__ X2, float* __restrict__ P) {
  const int b = blockIdx.x;
  const int c = threadIdx.x;
  if (c >= OUT2) return;
  const float* __restrict__ x = X2 + (size_t)b * NNODES * OUT2;

  float m8[8];
#pragma unroll
  for (int s = 0; s < 8; ++s) {
    float m = -3.402823466e38f;
    for (int i = 0; i < NNODES / 8; ++i)
      m = fmaxf(m, x[(size_t)(s * (NNODES / 8) + i) * OUT2 + c]);
    m8[s] = m;
  }
  float m4[4], m2[2];
#pragma unroll
  for (int i = 0; i < 4; ++i) m4[i] = fmaxf(m8[2 * i], m8[2 * i + 1]);
#pragma unroll
  for (int i = 0; i < 2; ++i) m2[i] = fmaxf(m4[2 * i], m4[2 * i + 1]);
  const float m1 = fmaxf(m2[0], m2[1]);

  float* __restrict__ p = P + (size_t)b * 1800;
  p[c] = m1;
#pragma unroll
  for (int i = 0; i < 2; ++i) p[120 + c * 2 + i] = m2[i];
#pragma unroll
  for (int i = 0; i < 4; ++i) p[360 + c * 4 + i] = m4[i];
#pragma unroll
  for (int i = 0; i < 8; ++i) p[840 + c * 8 + i] = m8[i];
}

// ---------------------------------------------------------------------------
// Kernel 5: final FC, out[b,o] = P[b,:] . Wfc[o,:] + bfc[o]  (32x104, tiny)
// ---------------------------------------------------------------------------
__global__ __launch_bounds__(256) void final_fc_kernel(
    const float* __restrict__ P, const float* __restrict__ Wfc,
    const float* __restrict__ bfc, float* __restrict__ out) {
  const int t = blockIdx.x * blockDim.x + threadIdx.x;
  if (t >= NB * LABEL) return;
  const int b = t / LABEL;
  const int o = t % LABEL;
  const float* __restrict__ p = P + (size_t)b * 1800;
  const float* __restrict__ w = Wfc + (size_t)o * 1800;
  float acc = bfc[o];
  for (int f = 0; f < 1800; ++f) acc += p[f] * w[f];
  out[(size_t)b * LABEL + o] = acc;
}

// ---------------------------------------------------------------------------
extern "C" void kernel_launch(void* const* d_in, const int* in_sizes, int n_in,
                              void* d_out, int out_size, void* d_ws,
                              size_t ws_size, hipStream_t stream) {
  const float* nodes = (const float*)d_in[0];   // [32,1024,256]
  const int* children = (const int*)d_in[1];    // [32,1024,8]
  const float* W1 = (const float*)d_in[2];      // [240,768]
  const float* b1 = (const float*)d_in[3];      // [240]
  const float* W2 = (const float*)d_in[4];      // [120,240]
  const float* b2 = (const float*)d_in[5];      // [120]
  const float* Wfc = (const float*)d_in[6];     // [104,1800]
  const float* bfc = (const float*)d_in[7];     // [104]
  float* out = (float*)d_out;                   // [32,104]

  char* ws = (char*)d_ws;
  const size_t x0_bytes = (size_t)MTOT * 768 * sizeof(float);   // 96 MB
  const size_t x1_bytes = (size_t)MTOT * OUT1 * sizeof(float);  // 30 MB
  float* X0 = (float*)ws;                       // [32768, 768]
  float* X1 = (float*)(ws + x0_bytes);          // [32768, 240]
  float* X2 = (float*)ws;                       // [32768, 120] (aliases X0)
  float* P = (float*)(ws + x0_bytes + x1_bytes);// [32, 1800]

  // 1) tree conv -> X0
  tree_conv_kernel<<<MTOT / 8, 256, 0, stream>>>(nodes, children, X0);

  // 2) X1 = tanh(X0 * W1^T + b1), N=240 -> 15 tiles
  gemm_bias_tanh_kernel<15><<<MTOT / 128, 256, 0, stream>>>(
      X0, W1, b1, X1, OUT1, 3 * EMB);

  // 3) X2 = tanh(X1 * W2^T + b2), N=120 -> 8 tiles (last half-masked)
  gemm_bias_tanh_kernel<8><<<MTOT / 128, 256, 0, stream>>>(
      X1, W2, b2, X2, OUT2, OUT1);

  // 4) pyramid pooling -> P
  pyramid_pool_kernel<<<NB, 128, 0, stream>>>(X2, P);

  // 5) final FC -> out
  final_fc_kernel<<<(NB * LABEL + 255) / 256, 256, 0, stream>>>(P, Wfc, bfc,
                                                                out);
}